// WaveletTransform_76416058131400
// MI455X (gfx1250) — compile-verified
//
#include <hip/hip_runtime.h>

typedef __attribute__((ext_vector_type(2))) float v2f;
typedef __attribute__((ext_vector_type(4))) float v4f;
typedef __attribute__((ext_vector_type(8))) float v8f;

// ---------------- compile-time banded-operator fragment table ----------------
// Per-lane tap selection tau = 4t + 2g - 2n is identical for the stage-1 B
// operand (n = output col) and the stage-2 A operand (n = output row).
// Bake all 9 K-step fragments for both filters into constant memory:
// OPTAB[t][lane] = { wL.x, wL.y, wH.x, wH.y }.
struct OpTab { float v[9][32][4]; };

constexpr float db4pick(const double* h, int t) {
    return (t >= 0 && t < 4) ? (float)h[t] : 0.0f;
}

constexpr OpTab make_optab() {
    OpTab o{};
    const double s3  = 1.7320508075688772935;
    const double nrm = 0.17677669529663688110;   // 1/(4*sqrt(2))
    const double h0[4] = {(1.0 + s3) * nrm, (3.0 + s3) * nrm,
                          (3.0 - s3) * nrm, (1.0 - s3) * nrm};
    const double h1[4] = {h0[3], -h0[2], h0[1], -h0[0]};  // h1[k] = (-1)^k h0[3-k]
    for (int t = 0; t < 9; ++t) {
        for (int lane = 0; lane < 32; ++lane) {
            int n = lane & 15, g = lane >> 4;
            int tau = 4 * t + 2 * g - 2 * n;
            o.v[t][lane][0] = db4pick(h0, tau);
            o.v[t][lane][1] = db4pick(h0, tau + 1);
            o.v[t][lane][2] = db4pick(h1, tau);
            o.v[t][lane][3] = db4pick(h1, tau + 1);
        }
    }
    return o;
}

__device__ __constant__ OpTab OPTAB = make_optab();

#define WMMA_F32(A, B, C) \
    __builtin_amdgcn_wmma_f32_16x16x4_f32(false, (A), false, (B), (short)0, (C), false, false)

// One-wave (wave32) workgroup computes a 16x16 tile of each subband (LL,LH,HL,HH)
// for one (b,c) image at one pyramid level. Fused separable db4 DWT with periodic
// boundary; both passes run on the matrix pipe via V_WMMA_F32_16X16X4_F32.
__global__ __launch_bounds__(32)
void dwt_db4_level(const float* __restrict__ src, int N, long long srcStride,
                   float* __restrict__ dstLL, int ldLL, long long llStride,
                   float* __restrict__ outDet, int M)
{
    const int lane = threadIdx.x;          // 0..31, wave32
    const int n    = lane & 15;            // fragment M/N index within half-wave
    const int g    = lane >> 4;            // lane-half selects K pair
    const int c0   = blockIdx.x * 16;      // subband col origin
    const int r0   = blockIdx.y * 16;      // subband row origin
    const int img  = blockIdx.z;

    src    += (long long)img * srcStride;
    dstLL  += (long long)img * llStride;
    outDet += (long long)img * (512LL * 512LL);

    // Input region: rows 2r0-1..2r0+32 (34), cols 2c0-1..2c0+34 (36), periodic wrap.
    __shared__ float Xs[34][38];   // stride 38 keeps float2 K-slabs 8B aligned
    __shared__ float Lr[36][17];   // horizontal-lo, rows = local input rows
    __shared__ float Hr[36][17];   // horizontal-hi

    // ---- staging: wrapped columns are per-lane invariants, row index is uniform ----
    int gcA = 2 * c0 - 1 + lane;       gcA += (gcA < 0) ? N : 0;  gcA -= (gcA >= N) ? N : 0;
    int gcB = 2 * c0 + 31 + lane;      gcB -= (gcB >= N) ? N : 0;   // lanes 0..3 only
    int gr  = 2 * r0 - 1;              gr  += (gr  < 0) ? N : 0;
    for (int r = 0; r < 34; ++r) {
        const float* p = src + (long long)gr * N;
        Xs[r][lane] = p[gcA];
        if (lane < 4) Xs[r][32 + lane] = p[gcB];
        ++gr;  gr = (gr >= N) ? 0 : gr;
    }
    if (lane < 16) {  // zero the K-tail rows so 0-weight taps never hit garbage
        Lr[34][lane] = 0.0f; Lr[35][lane] = 0.0f;
        Hr[34][lane] = 0.0f; Hr[35][lane] = 0.0f;
    }

    // ---- operator fragments: 9 coalesced b128 loads, shared by all 90 WMMAs ----
    v2f wL[9], wH[9];
    #pragma unroll
    for (int t = 0; t < 9; ++t) {
        v4f w = *(const v4f*)OPTAB.v[t][lane];
        wL[t].x = w.x;  wL[t].y = w.y;
        wH[t].x = w.z;  wH[t].y = w.w;
    }
    __syncthreads();

    // ---------- Stage 1: horizontal pass (A = data, B = operator) ----------
    // Lr[row][n] = sum_j Xs[row][j] * h0[j - 2n]
    const int chunkR[3] = {0, 16, 18};     // covers local rows 0..33
    #pragma unroll
    for (int ch = 0; ch < 3; ++ch) {
        const int R = chunkR[ch];
        v8f accL = {0.f,0.f,0.f,0.f,0.f,0.f,0.f,0.f};
        v8f accH = {0.f,0.f,0.f,0.f,0.f,0.f,0.f,0.f};
        #pragma unroll
        for (int t = 0; t < 9; ++t) {
            // A fragment: lane holds row M=n, K = 2g (.x) / 2g+1 (.y)
            v2f a = *(const v2f*)&Xs[R + n][4 * t + 2 * g];
            accL = WMMA_F32(a, wL[t], accL);
            accH = WMMA_F32(a, wH[t], accH);
        }
        // C layout: value (M = v + 8g, N = n) in acc[v]
        #pragma unroll
        for (int v = 0; v < 8; ++v) {
            Lr[R + v + 8 * g][n] = accL[v];
            Hr[R + v + 8 * g][n] = accH[v];
        }
    }
    __syncthreads();

    // ---------- Stage 2: vertical pass (A = operator, B = data) ----------
    // LL[m][n] = sum_j h0[j - 2m] * Lr[j][n], etc.
    v8f LL = {0.f,0.f,0.f,0.f,0.f,0.f,0.f,0.f};
    v8f LH = {0.f,0.f,0.f,0.f,0.f,0.f,0.f,0.f};
    v8f HL = {0.f,0.f,0.f,0.f,0.f,0.f,0.f,0.f};
    v8f HH = {0.f,0.f,0.f,0.f,0.f,0.f,0.f,0.f};
    #pragma unroll
    for (int t = 0; t < 9; ++t) {
        int rr = 4 * t + 2 * g;            // B: lane holds col N=n, K rows rr, rr+1
        v2f bLr, bHr;
        bLr.x = Lr[rr][n];  bLr.y = Lr[rr + 1][n];
        bHr.x = Hr[rr][n];  bHr.y = Hr[rr + 1][n];
        LL = WMMA_F32(wL[t], bLr, LL);   // v-lo of h-lo
        LH = WMMA_F32(wH[t], bLr, LH);   // v-hi of h-lo  -> top-right
        HL = WMMA_F32(wL[t], bHr, HL);   // v-lo of h-hi  -> bottom-left
        HH = WMMA_F32(wH[t], bHr, HH);   // v-hi of h-hi  -> bottom-right
    }

    // ---------- Store ----------
    // Detail subbands are write-once/never-re-read: stream them (non-temporal)
    // so the LL chain (re-read by the next level) keeps the 192MB L2.
    #pragma unroll
    for (int v = 0; v < 8; ++v) {
        int orow = r0 + v + 8 * g;
        int ocol = c0 + n;
        dstLL[(long long)orow * ldLL + ocol] = LL[v];
        __builtin_nontemporal_store(LH[v], &outDet[(long long)orow * 512 + (ocol + M)]);
        __builtin_nontemporal_store(HL[v], &outDet[(long long)(orow + M) * 512 + ocol]);
        __builtin_nontemporal_store(HH[v], &outDet[(long long)(orow + M) * 512 + (ocol + M)]);
    }
}

extern "C" void kernel_launch(void* const* d_in, const int* in_sizes, int n_in,
                              void* d_out, int out_size, void* d_ws, size_t ws_size,
                              hipStream_t stream) {
    const float* x = (const float*)d_in[0];
    float* out = (float*)d_out;
    const int BC = in_sizes[0] / (512 * 512);          // 8*16 = 128 images

    float* LL1 = (float*)d_ws;                          // BC * 256 * 256 floats
    float* LL2 = LL1 + (long long)BC * 256 * 256;       // BC * 128 * 128 floats

    dim3 blk(32, 1, 1);
    // Level 1: 512 -> 256 subbands; LL1 to workspace, details to out quadrants.
    dwt_db4_level<<<dim3(16, 16, BC), blk, 0, stream>>>(
        x,   512, 512LL * 512, LL1, 256, 256LL * 256, out, 256);
    // Level 2: 256 -> 128
    dwt_db4_level<<<dim3(8, 8, BC), blk, 0, stream>>>(
        LL1, 256, 256LL * 256, LL2, 128, 128LL * 128, out, 128);
    // Level 3: 128 -> 64; LL3 goes to out top-left (ld 512)
    dwt_db4_level<<<dim3(4, 4, BC), blk, 0, stream>>>(
        LL2, 128, 128LL * 128, out, 512, 512LL * 512, out, 64);
}